// TextLevelGCN_30382598652337
// MI455X (gfx1250) — compile-verified
//
#include <hip/hip_runtime.h>

#define N_NODES 65536
#define N_EDGES 524288
#define DIMS 128
#define N_GRAPHS 512
#define N_CLASSES 20

typedef __attribute__((ext_vector_type(2))) float v2f;
typedef __attribute__((ext_vector_type(8))) float v8f;

// Order-preserving float <-> uint encoding so integer atomic max == float max.
__device__ __forceinline__ unsigned encf(float f) {
  unsigned u = __float_as_uint(f);
  return (u & 0x80000000u) ? ~u : (u | 0x80000000u);
}
__device__ __forceinline__ float decf(unsigned e) {
  unsigned u = (e & 0x80000000u) ? (e & 0x7FFFFFFFu) : ~e;
  return __uint_as_float(u);
}
#define ENC_NEG_INF 0x007FFFFFu  // encf(-inf)

// ---------------------------------------------------------------- init
__global__ __launch_bounds__(256) void k_init(unsigned* __restrict__ agg,
                                              int* __restrict__ deg,
                                              float* __restrict__ gf) {
  size_t stride = (size_t)gridDim.x * blockDim.x;
  size_t total = (size_t)N_NODES * DIMS;
  for (size_t idx = (size_t)blockIdx.x * blockDim.x + threadIdx.x; idx < total;
       idx += stride) {
    agg[idx] = ENC_NEG_INF;
    if (idx < (size_t)N_NODES) deg[idx] = 0;
    if (idx < (size_t)N_GRAPHS * DIMS) gf[idx] = 0.0f;
  }
}

// ----------------------------------------------- edge scatter (wave32/edge)
__global__ __launch_bounds__(256) void k_edges(const float* __restrict__ h,
                                               const float* __restrict__ w,
                                               const int* __restrict__ src,
                                               const int* __restrict__ dst,
                                               unsigned* __restrict__ agg,
                                               int* __restrict__ deg) {
  int gid = blockIdx.x * blockDim.x + threadIdx.x;
  int e = gid >> 5;
  int lane = gid & 31;
  if (e >= N_EDGES) return;
  int s = src[e];
  int d = dst[e];
  float we = w[e];
  // 32 lanes x float4 = 128 dims, coalesced 512B gather (L2-resident h)
  float4 hv = ((const float4*)(h + (size_t)s * DIMS))[lane];
  unsigned* ap = agg + (size_t)d * DIMS + lane * 4;
  atomicMax(ap + 0, encf(hv.x * we));
  atomicMax(ap + 1, encf(hv.y * we));
  atomicMax(ap + 2, encf(hv.z * we));
  atomicMax(ap + 3, encf(hv.w * we));
  if (lane == 0) atomicAdd(deg + d, 1);
}

// --------------------------------------- node reduce -> per-graph sums
__global__ __launch_bounds__(256) void k_nodes(const unsigned* __restrict__ agg,
                                               const int* __restrict__ deg,
                                               const int* __restrict__ gids,
                                               float* __restrict__ gf) {
  int gid = blockIdx.x * blockDim.x + threadIdx.x;
  int n = gid >> 5;
  int lane = gid & 31;
  if (n >= N_NODES) return;
  int dg = deg[n];
  int g = gids[n];
  uint4 av = ((const uint4*)(agg + (size_t)n * DIMS))[lane];
  float x0 = (dg > 0) ? decf(av.x) : 0.0f;
  float x1 = (dg > 0) ? decf(av.y) : 0.0f;
  float x2 = (dg > 0) ? decf(av.z) : 0.0f;
  float x3 = (dg > 0) ? decf(av.w) : 0.0f;
  float* gp = gf + (size_t)g * DIMS + lane * 4;
  atomicAdd(gp + 0, x0);
  atomicAdd(gp + 1, x1);
  atomicAdd(gp + 2, x2);
  atomicAdd(gp + 3, x3);
}

// ----------------------- classifier: relu(gf) @ W.T + b via f32 WMMA
// One wave computes one 16x16 tile of out[512,20]; K=128 via 32x V_WMMA_F32_16X16X4_F32.
__global__ __launch_bounds__(32) void k_classify(const float* __restrict__ gf,
                                                 const float* __restrict__ Wm,
                                                 const float* __restrict__ bias,
                                                 float* __restrict__ out) {
  int lane = threadIdx.x;   // 0..31 (wave32)
  int tm = blockIdx.x;      // 32 M-tiles over 512 rows
  int tn = blockIdx.y;      // 2 N-tiles over 20 cols (padded to 32)
  int half = lane >> 4;     // 0: K={0,1}, 1: K={2,3} (A layout per ISA 16x4 table)
  int l15 = lane & 15;
  int mrow = tm * 16 + l15;
  int ncol = tn * 16 + l15;
  int valid = (ncol < N_CLASSES) ? 1 : 0;
  const float* arow = gf + (size_t)mrow * DIMS;
  const float* brow = Wm + (size_t)(valid ? ncol : 0) * DIMS;  // clamp, no divergence
  float bmask = valid ? 1.0f : 0.0f;

  v8f acc = {};
#pragma unroll
  for (int k0 = 0; k0 < DIMS; k0 += 4) {
    int kb = k0 + (half << 1);
    v2f a, bb;
    a.x = fmaxf(arow[kb], 0.0f);       // fused ReLU on A operand
    a.y = fmaxf(arow[kb + 1], 0.0f);
    bb.x = brow[kb] * bmask;           // B[k][n] = W[n][k], zero-masked n>=20
    bb.y = brow[kb + 1] * bmask;
    acc = __builtin_amdgcn_wmma_f32_16x16x4_f32(
        /*neg_a=*/false, a, /*neg_b=*/false, bb,
        /*c_mod=*/(short)0, acc, /*reuse_a=*/false, /*reuse_b=*/false);
  }

  float bv = valid ? bias[ncol] : 0.0f;
#pragma unroll
  for (int r = 0; r < 8; ++r) {
    int m = tm * 16 + r + (half << 3);  // C/D layout: lanes16-31 -> M=r+8
    if (valid) out[(size_t)m * N_CLASSES + ncol] = acc[r] + bv;
  }
}

extern "C" void kernel_launch(void* const* d_in, const int* in_sizes, int n_in,
                              void* d_out, int out_size, void* d_ws, size_t ws_size,
                              hipStream_t stream) {
  const float* h = (const float*)d_in[0];
  const float* w = (const float*)d_in[1];
  const int* src = (const int*)d_in[2];
  const int* dst = (const int*)d_in[3];
  const int* gids = (const int*)d_in[4];
  const float* Wm = (const float*)d_in[5];
  const float* b = (const float*)d_in[6];
  float* out = (float*)d_out;

  // workspace layout: [agg_enc: N*D u32][deg: N i32][gf: G*D f32] = ~32.5 MB
  unsigned* agg = (unsigned*)d_ws;
  int* deg = (int*)((char*)d_ws + (size_t)N_NODES * DIMS * 4);
  float* gf = (float*)((char*)d_ws + (size_t)N_NODES * DIMS * 4 + (size_t)N_NODES * 4);

  k_init<<<2048, 256, 0, stream>>>(agg, deg, gf);
  k_edges<<<(N_EDGES * 32) / 256, 256, 0, stream>>>(h, w, src, dst, agg, deg);
  k_nodes<<<(N_NODES * 32) / 256, 256, 0, stream>>>(agg, deg, gids, gf);
  dim3 grid_c(32, 2);
  k_classify<<<grid_c, 32, 0, stream>>>(gf, Wm, b, out);
}